// SparseLinear_63771674411297
// MI455X (gfx1250) — compile-verified
//
#include <hip/hip_runtime.h>

typedef float v2f __attribute__((ext_vector_type(2)));
typedef float v8f __attribute__((ext_vector_type(8)));

#define MK 4096
#define NN 4096
#define TM 32      // workgroup M tile == mask block size -> uniform skip
#define TN 256     // workgroup N tile (8 waves x 32)
#define TK 32      // K step == mask block size
#define LDA 36     // padded LDS strides: 36*4B = 144B (16B aligned, conflict-free)
#define LDB 36

__global__ __launch_bounds__(256)
void sparse_linear_sdd_wmma(const float* __restrict__ A,      // [M,K] input
                            const float* __restrict__ W,      // [N,K] weight
                            const float* __restrict__ bias,   // [N]
                            const int*   __restrict__ mask,   // [M,K] block-constant
                            float* __restrict__ out)          // [M,N]
{
    __shared__ float ldsA[TM * LDA];
    __shared__ float ldsB[TN * LDB];

    const int tid   = threadIdx.x;
    const int wave  = tid >> 5;
    const int lane  = tid & 31;
    const int l16   = lane & 15;
    const int khalf = (lane >> 4) * 2;          // 0 for lanes 0-15, 2 for 16-31

    const int m_base = blockIdx.x * TM;         // one 32-row mask block
    const int n_base = blockIdx.y * TN;
    const int n_wave = wave * 32;               // wave's 32 columns inside tile

    v8f acc[2][2];
#pragma unroll
    for (int mi = 0; mi < 2; ++mi)
#pragma unroll
        for (int ni = 0; ni < 2; ++ni)
            acc[mi][ni] = (v8f)(0.0f);

    for (int kb = 0; kb < MK / TK; ++kb) {
        // Mask is constant over each 32x32 block: probe one element.
        // Address is uniform (blockIdx + loop var) -> scalar load, and the
        // branch is workgroup-uniform so barriers below stay legal.
        if (mask[(size_t)m_base * MK + (size_t)kb * TK] == 0) continue;

        { // stage A tile 32x32: 256 threads x one float4
            const int m  = tid >> 3;
            const int kq = tid & 7;
            const float4 va = *(const float4*)(A + (size_t)(m_base + m) * MK + kb * TK + kq * 4);
            *(float4*)(&ldsA[m * LDA + kq * 4]) = va;
        }
#pragma unroll
        for (int j = 0; j < 8; ++j) { // stage B tile 256x32 (n-major, k contiguous)
            const int idx = tid + j * 256;
            const int n   = idx >> 3;
            const int kq  = idx & 7;
            const float4 vb = *(const float4*)(W + (size_t)(n_base + n) * MK + kb * TK + kq * 4);
            *(float4*)(&ldsB[n * LDB + kq * 4]) = vb;
        }
        __syncthreads();

#pragma unroll
        for (int k = 0; k < TK; k += 4) {
            v2f afrag[2], bfrag[2];
#pragma unroll
            for (int mi = 0; mi < 2; ++mi)   // A 16x4: lane=M row, K across halves
                afrag[mi] = *(const v2f*)(&ldsA[(mi * 16 + l16) * LDA + k + khalf]);
#pragma unroll
            for (int ni = 0; ni < 2; ++ni)   // B 4x16: lane=N col, K across halves
                bfrag[ni] = *(const v2f*)(&ldsB[(n_wave + ni * 16 + l16) * LDB + k + khalf]);
#pragma unroll
            for (int mi = 0; mi < 2; ++mi)
#pragma unroll
                for (int ni = 0; ni < 2; ++ni)
                    acc[mi][ni] = __builtin_amdgcn_wmma_f32_16x16x4_f32(
                        false, afrag[mi], false, bfrag[ni],
                        (short)0, acc[mi][ni], false, false);
        }
        __syncthreads();
    }

    // Epilogue: add bias, store fp32. C/D layout: VGPR i -> M = i + 8*(lane>=16).
#pragma unroll
    for (int ni = 0; ni < 2; ++ni) {
        const int col = n_base + n_wave + ni * 16 + l16;
        const float bv = bias[col];
#pragma unroll
        for (int mi = 0; mi < 2; ++mi) {
#pragma unroll
            for (int i = 0; i < 8; ++i) {
                const int row = m_base + mi * 16 + i + (lane >> 4) * 8;
                out[(size_t)row * NN + col] = acc[mi][ni][i] + bv;
            }
        }
    }
}

extern "C" void kernel_launch(void* const* d_in, const int* in_sizes, int n_in,
                              void* d_out, int out_size, void* d_ws, size_t ws_size,
                              hipStream_t stream) {
    const float* A    = (const float*)d_in[0];   // input  [4096,4096] f32
    const float* W    = (const float*)d_in[1];   // weight [4096,4096] f32 (N,K)
    const float* bias = (const float*)d_in[2];   // bias   [4096] f32
    const int*   mask = (const int*)d_in[3];     // mask   [4096,4096] i32
    float* out = (float*)d_out;

    dim3 grid(MK / TM, NN / TN);  // (128, 16)
    dim3 block(256);
    sparse_linear_sdd_wmma<<<grid, block, 0, stream>>>(A, W, bias, mask, out);
}